// Model_26422638805319
// MI455X (gfx1250) — compile-verified
//
#include <hip/hip_runtime.h>
#include <hip/hip_bf16.h>
#include <math.h>

// ---------------------------------------------------------------------------
// Fused proposal-scoring kernel for MI455X (gfx1250, wave32).
//
// score(start,end) = (csum[end]-csum[start])·sent / (l+1e-8)
//                  = ( Σ_{t in [start,end)} video[t]·sent ) / (l+1e-8)
// so we only need per-frame dots d[t] (a GEMV), done with f32 WMMA 16x16x4:
//   A (16x4)  = sentence chunk s[c..c+3] broadcast into all 16 rows
//   B (4x16)  = video[frame n, c..c+3] for 16 frames n
//   C (16x16) = every column j holds d[frame j]  (read from C VGPR0, lanes 0..15)
// Fragment layout (ISA 7.12.2): lanes 0-15 hold K={0,1}, lanes 16-31 K={2,3}.
// ---------------------------------------------------------------------------

typedef __attribute__((ext_vector_type(2))) float v2f;
typedef __attribute__((ext_vector_type(8))) float v8f;

#define NSLOT 37
#define LFRM  2048
#define DDIM  256
#define NTOPK 10

// Static slot tables for L=2048 (Python round-half-to-even semantics),
// enumerated in (moment_factor, overlap_factor, k) order like the reference.
__device__ __constant__ int c_l[NSLOT] = {
  205,205,205,205,205,205,205,
  410,410,410,410,410,410,410,410,
  614,614,614,614,614,614,614,
  819,819,819,819,819,819,819,
  1024,1024,1024,1024,1024,1024,1024,1024};
__device__ __constant__ int c_lr[NSLOT] = {
  103,103,51,51,51,51,51,
  205,205,205,102,102,102,102,102,
  307,307,307,154,154,154,154,
  409,409,409,205,205,205,205,
  512,512,512,256,256,256,256,256};
__device__ __constant__ int c_k[NSLOT] = {
  0,1,0,1,2,3,4,
  0,1,2,0,1,2,3,4,
  0,1,2,0,1,2,3,
  0,1,2,0,1,2,3,
  0,1,2,0,1,2,3,4};

__global__ __launch_bounds__(256) void proposal_topk_kernel(
    const float* __restrict__ sent,   // [B, 256]
    const float* __restrict__ video,  // [B, 2048, 256]
    const float* __restrict__ attn,   // [B, 2048]
    float* __restrict__ out,          // [B*10*2 props][B*10 scores]
    int B)
{
  __shared__ __align__(16) float s_lds[DDIM];
  __shared__ __align__(16) float d_lds[LFRM];
  __shared__ unsigned long long amax_key;
  __shared__ float sl_score[NSLOT];
  __shared__ float sl_final[NSLOT];
  __shared__ int   sl_start[NSLOT];
  __shared__ int   sl_end[NSLOT];
  __shared__ int   sl_valid[NSLOT];

  const int b    = blockIdx.x;
  const int tid  = threadIdx.x;      // 256 threads = 8 waves (wave32)
  const int wave = tid >> 5;
  const int lane = tid & 31;

  // Stage sentence vector in LDS (broadcast source for WMMA A fragments).
  s_lds[tid] = sent[(size_t)b * DDIM + tid];
  if (tid == 0) amax_key = 0ull;
  __syncthreads();

  // ---- Phase 1: per-frame dots via V_WMMA_F32_16X16X4_F32 -----------------
  const int n   = lane & 15;           // frame-within-tile (B-matrix column)
  const int off = (lane >> 4) << 1;    // K offset: lanes 0-15 -> 0, 16-31 -> 2

#if __has_builtin(__builtin_amdgcn_wmma_f32_16x16x4_f32)
  for (int it = 0; it < 8; ++it) {
    const int fb0 = (wave * 2 + it * 16) * 16;   // 128 tiles of 16 frames
    const int fb1 = fb0 + 16;
    const float* r0 = video + ((size_t)b * LFRM + fb0 + n) * DDIM + off;
    const float* r1 = video + ((size_t)b * LFRM + fb1 + n) * DDIM + off;
    if (it < 7) {  // speculative prefetch of next tile pair (global_prefetch_b8)
      __builtin_prefetch(r0 + 256 * DDIM, 0, 0);
      __builtin_prefetch(r1 + 256 * DDIM, 0, 0);
    }
    v8f acc0 = {};
    v8f acc1 = {};
#pragma unroll 4
    for (int c = 0; c < DDIM; c += 4) {
      v2f af = *(const v2f*)(s_lds + c + off);   // ds_load_b64 (broadcast)
      v2f b0 = *(const v2f*)(r0 + c);            // global_load_b64
      v2f b1 = *(const v2f*)(r1 + c);
      acc0 = __builtin_amdgcn_wmma_f32_16x16x4_f32(
          false, af, false, b0, (short)0, acc0, false, false);
      acc1 = __builtin_amdgcn_wmma_f32_16x16x4_f32(
          false, af, false, b1, (short)0, acc1, false, false);
    }
    // C VGPR0, lanes 0-15: (M=0, N=lane) -> d[frame fb+lane]
    if (lane < 16) {
      d_lds[fb0 + n] = acc0[0];
      d_lds[fb1 + n] = acc1[0];
    }
  }
#else
  for (int t = tid; t < LFRM; t += 256) {
    const float* r = video + ((size_t)b * LFRM + t) * DDIM;
    float s = 0.f;
    for (int c = 0; c < DDIM; ++c) s = fmaf(r[c], s_lds[c], s);
    d_lds[t] = s;
  }
#endif

  // ---- Phase 2: argmax(attn), first-max tie-break (packed 64-bit keys) ----
  {
    unsigned long long k = 0ull;
    for (int t = tid; t < LFRM; t += 256) {
      unsigned fb = __float_as_uint(attn[(size_t)b * LFRM + t]); // attn >= 0
      unsigned long long key =
          ((unsigned long long)fb << 32) | (unsigned)(0x7FFFFFFF - t);
      if (key > k) k = key;
    }
    atomicMax(&amax_key, k);   // ds_max_u64
  }
  __syncthreads();
  const int anchor = 0x7FFFFFFF - (int)(unsigned)(amax_key & 0xFFFFFFFFull);

  // ---- Phase 3: slot window sums (one wave per slot, round-robin) ---------
  for (int s = wave; s < NSLOT; s += 8) {
    const int l  = c_l[s];
    const int lr = c_lr[s];
    const int kk = c_k[s];
    const int mi = (anchor - l > 0) ? (anchor - l) : 0;
    const int st = mi + kk * lr;
    const int en = st + l;
    const int hi = (anchor + l < LFRM) ? (anchor + l) : LFRM;
    const bool valid = (st <= anchor) && (en <= hi);
    float sum = 0.f;
    if (valid) {
      for (int i = st + lane; i < en; i += 32) sum += d_lds[i];
    }
    // wave32 butterfly reduction
    for (int o = 16; o > 0; o >>= 1) sum += __shfl_xor(sum, o, 32);
    if (lane == 0) {
      sl_start[s] = st;
      sl_end[s]   = en;
      sl_valid[s] = valid ? 1 : 0;
      sl_score[s] = valid ? sum / ((float)l + 1e-8f) : -INFINITY;
    }
  }
  __syncthreads();

  // ---- Phase 4: dedup identical (start,end) among valid slots -------------
  if (tid < NSLOT) {
    float sc = sl_score[tid];
    if (sl_valid[tid]) {
      for (int j = 0; j < tid; ++j) {
        if (sl_valid[j] && sl_start[j] == sl_start[tid] &&
            sl_end[j] == sl_end[tid]) { sc = -INFINITY; break; }
      }
    }
    sl_final[tid] = sc;
  }
  __syncthreads();

  // ---- Phase 5: stable top-10 (lowest index wins ties, like lax.top_k) ----
  if (tid == 0) {
    unsigned long long taken = 0ull;
    float* props  = out + (size_t)b * (NTOPK * 2);
    float* scores = out + (size_t)B * NTOPK * 2 + (size_t)b * NTOPK;
    const float invL = 1.0f / (float)LFRM;
    for (int r = 0; r < NTOPK; ++r) {
      int bi = -1;
      float best = -INFINITY;
      for (int i = 0; i < NSLOT; ++i) {
        if (!((taken >> i) & 1ull) && (bi < 0 || sl_final[i] > best)) {
          bi = i;
          best = sl_final[i];
        }
      }
      taken |= (1ull << bi);
      props[r * 2 + 0] = (float)sl_start[bi] * invL;
      props[r * 2 + 1] = (float)sl_end[bi] * invL;
      scores[r] = best;
    }
  }
}

extern "C" void kernel_launch(void* const* d_in, const int* in_sizes, int n_in,
                              void* d_out, int out_size, void* d_ws, size_t ws_size,
                              hipStream_t stream) {
  (void)n_in; (void)out_size; (void)d_ws; (void)ws_size;
  const float* sent  = (const float*)d_in[0];  // [B,256]
  const float* video = (const float*)d_in[1];  // [B,2048,256]
  const float* attn  = (const float*)d_in[2];  // [B,2048]
  float* out = (float*)d_out;                  // [B*10*2] props ++ [B*10] scores
  const int B = in_sizes[2] / LFRM;            // 64
  proposal_topk_kernel<<<dim3(B), dim3(256), 0, stream>>>(sent, video, attn, out, B);
}